// D4_codebook_8512625180848
// MI455X (gfx1250) — compile-verified
//
#include <hip/hip_runtime.h>
#include <stdint.h>

typedef __attribute__((ext_vector_type(2))) float v2f;
typedef __attribute__((ext_vector_type(8))) float v8f;

#define BLOCK 256
#define WAVES_PER_BLOCK (BLOCK / 32)

// D4 nearest-codeword quantization via V_WMMA_F32_16X16X4_F32.
// score[m][n] = 2<X[m], grid[n]> - |grid[n]|^2.
// D = A(16x4 X tile) * B(4x16 of 2*grid^T) + 0 (inline-0 C); the -|g|^2 bias is
// one VALU op per candidate in the argmax fold. A-tile loads are software-
// pipelined one iteration ahead plus a global_prefetch_b8 two ahead, so the
// s_wait_loadcnt before the WMMA block has a full loop body of latency cover.
__global__ __launch_bounds__(BLOCK) void d4_quant_wmma(
    const float* __restrict__ X,         // [N,4]
    const float* __restrict__ grid,      // [256,4]
    const float* __restrict__ grid_norm, // [256]
    float* __restrict__ outVec,          // [N,4]
    float* __restrict__ outIdx,          // [N] (index value as float)
    int ntiles)                          // N / 16
{
    const int lane  = threadIdx.x & 31;
    const int laneN = lane & 15;   // codeword-in-group (B/C/D N index), row for A load
    const int hi    = lane >> 4;   // which K-half this lane supplies
    const int k0    = hi * 2;

    // ---- Preload B (= 2*grid^T, per 16-codeword group) and -|g|^2 ----
    // B layout (4x16 f32, 2 VGPRs): v0: lanes0-15 K=0 / lanes16-31 K=2;
    //                               v1: lanes0-15 K=1 / lanes16-31 K=3; N = lane&15.
    v2f   Breg[16];
    float negnorm[16];
#pragma unroll
    for (int g = 0; g < 16; ++g) {
        const int n = g * 16 + laneN;
        v2f gb = *(const v2f*)(grid + n * 4 + k0);
        Breg[g].x = 2.0f * gb.x;
        Breg[g].y = 2.0f * gb.y;
        negnorm[g] = -grid_norm[n];
    }

    const v8f zeroC = {};  // SRC2 inline 0

    const int gwave  = blockIdx.x * WAVES_PER_BLOCK + (threadIdx.x >> 5);
    const int nwaves = gridDim.x * WAVES_PER_BLOCK;

    // A layout (16x4 f32, 2 VGPRs): lanes0-15 hold K=0,1; lanes16-31 hold K=2,3.
    const int t0 = (gwave < ntiles) ? gwave : 0;   // wave-uniform clamp
    v2f a = *(const v2f*)(X + (size_t)(t0 * 16 + laneN) * 4 + k0);

    for (int tile = gwave; tile < ntiles; tile += nwaves) {
        const int m0 = tile * 16;

        // Issue next iteration's A load now (consumed next iteration).
        const int tnext  = tile + nwaves;
        const int tclamp = (tnext < ntiles) ? tnext : tile;     // uniform
        v2f a_next = *(const v2f*)(X + (size_t)(tclamp * 16 + laneN) * 4 + k0);

        // Prefetch two iterations ahead (global_prefetch_b8: no counters/VGPRs).
        const int tpf = tnext + nwaves;
        if (tpf < ntiles)                                        // uniform
            __builtin_prefetch(X + (size_t)(tpf * 16 + laneN) * 4 + k0, 0, 1);

        v8f best;
        int idx[8];

        // Group 0 initializes the running argmax directly.
        {
            v8f d0 = __builtin_amdgcn_wmma_f32_16x16x4_f32(
                false, a, false, Breg[0], (short)0, zeroC, false, false);
#pragma unroll
            for (int j = 0; j < 8; ++j) { best[j] = d0[j] + negnorm[0]; idx[j] = laneN; }
        }

#pragma unroll
        for (int g = 1; g < 16; ++g) {
            v8f d = __builtin_amdgcn_wmma_f32_16x16x4_f32(
                false, a, false, Breg[g], (short)0, zeroC, false, false);
            const int n = g * 16 + laneN;
#pragma unroll
            for (int j = 0; j < 8; ++j) {
                const float sc = d[j] + negnorm[g];  // full score: 2<x,g> - |g|^2
                const bool  gt = sc > best[j];       // strict > keeps first index
                best[j] = gt ? sc : best[j];
                idx[j]  = gt ? n  : idx[j];
            }
        }

        // ---- Reduce across the 16 lanes of each half-wave ----
        // C/D layout: lanes0-15 -> M=j, lanes16-31 -> M=j+8; N = lane&15.
#pragma unroll
        for (int m = 8; m >= 1; m >>= 1) {
#pragma unroll
            for (int j = 0; j < 8; ++j) {
                const float ov = __shfl_xor(best[j], m, 32);
                const int   oi = __shfl_xor(idx[j],  m, 32);
                const bool  gt = (ov > best[j]) || (ov == best[j] && oi < idx[j]);
                best[j] = gt ? ov : best[j];
                idx[j]  = gt ? oi : idx[j];
            }
        }

        // Lane 0 holds results for M = m0..m0+7, lane 16 for M = m0+8..m0+15.
        if (laneN == 0) {
            const int mb = m0 + hi * 8;
#pragma unroll
            for (int j = 0; j < 8; ++j) {
                const int id = idx[j];
                ((float4*)outVec)[mb + j] = ((const float4*)grid)[id];
                outIdx[mb + j] = (float)id;
            }
        }

        a = a_next;
    }
}

extern "C" void kernel_launch(void* const* d_in, const int* in_sizes, int n_in,
                              void* d_out, int out_size, void* d_ws, size_t ws_size,
                              hipStream_t stream) {
    const float* X         = (const float*)d_in[0];
    const float* grid      = (const float*)d_in[1];
    const float* grid_norm = (const float*)d_in[2];

    const int N      = in_sizes[0] / 4;   // number of 4-d vectors
    const int ntiles = N / 16;            // 16 vectors per WMMA M-tile

    float* outVec = (float*)d_out;                   // [N*4]
    float* outIdx = (float*)d_out + (size_t)N * 4;   // [N]

    // 512 blocks * 8 waves = 4096 waves; 65536 tiles -> 16 tiles/wave, uniform,
    // amortizing the per-wave codebook preload and filling ~4 waves/SIMD.
    d4_quant_wmma<<<dim3(512), dim3(BLOCK), 0, stream>>>(
        X, grid, grid_norm, outVec, outIdx, ntiles);
}